// wilson_6459630813548
// MI455X (gfx1250) — compile-verified
//
#include <hip/hip_runtime.h>
#include <stdint.h>

// Wilson-Dirac dslash, 32^4 lattice, complex64 (float2).
// src layout:  [spin=4][color=3][t][z][y][x]  -> plane stride 2^20 complex
// U   layout:  [row=3][col=3][mu=4][t][z][y][x]
// dest = src - 0.1 * sum_mu [ (I-g_mu) U_mu(x) src(x+mu) + (I+g_mu) U_mu^dag(x-mu) src(x-mu) ]

#define L4_SHIFT 20
static constexpr unsigned NBLK = (1u << L4_SHIFT) / 256u; // 4096 blocks
static constexpr float MKAPPA = -0.1f;                    // -kappa/u0

__device__ __forceinline__ float2 cadd(float2 a, float2 b) { return make_float2(a.x + b.x, a.y + b.y); }
__device__ __forceinline__ float2 csub(float2 a, float2 b) { return make_float2(a.x - b.x, a.y - b.y); }
__device__ __forceinline__ float2 addi(float2 a, float2 b) { return make_float2(a.x - b.y, a.y + b.x); } // a + i*b
__device__ __forceinline__ float2 subi(float2 a, float2 b) { return make_float2(a.x + b.y, a.y - b.x); } // a - i*b
__device__ __forceinline__ float2 muli(float2 a)  { return make_float2(-a.y, a.x); }  //  i*a
__device__ __forceinline__ float2 mulmi(float2 a) { return make_float2(a.y, -a.x); }  // -i*a

__device__ __forceinline__ void cfma(float2& acc, float2 u, float2 v) { // acc += u*v
    acc.x = fmaf(u.x, v.x, fmaf(-u.y, v.y, acc.x));
    acc.y = fmaf(u.x, v.y, fmaf( u.y, v.x, acc.y));
}
__device__ __forceinline__ void cfma_cj(float2& acc, float2 u, float2 v) { // acc += conj(u)*v
    acc.x = fmaf(u.x, v.x, fmaf( u.y, v.y, acc.x));
    acc.y = fmaf(u.x, v.y, fmaf(-u.y, v.x, acc.y));
}

// U_mu(n) row-major: Umat[C*3+c] = U[C][c][mu](n)
__device__ __forceinline__ void loadU(const float2* __restrict__ U2, int mu, unsigned n, float2 Umat[9]) {
#pragma unroll
    for (int e = 0; e < 9; ++e)
        Umat[e] = U2[((unsigned)(e * 4 + mu) << L4_SHIFT) + n];
}
// For the dagger term store Umat[C*3+c] = U[c][C][mu](n); conj applied in cfma_cj.
__device__ __forceinline__ void loadUdag(const float2* __restrict__ U2, int mu, unsigned n, float2 Umat[9]) {
#pragma unroll
    for (int C = 0; C < 3; ++C)
#pragma unroll
        for (int c = 0; c < 3; ++c)
            Umat[C * 3 + c] = U2[((unsigned)((c * 3 + C) * 4 + mu) << L4_SHIFT) + n];
}

__device__ __forceinline__ void loadPsiG(const float2* __restrict__ s2, unsigned n, float2 psi[12]) {
#pragma unroll
    for (int k = 0; k < 12; ++k) psi[k] = s2[((unsigned)k << L4_SHIFT) + n];
}
__device__ __forceinline__ void loadPsiL(const float2 (*sp)[256], int lt, float2 psi[12]) {
#pragma unroll
    for (int k = 0; k < 12; ++k) psi[k] = sp[k][lt];
}

// One hopping term: spin-project (half-spinor), SU(3) multiply, reconstruct, accumulate.
// PLUS==0: (I - g_mu) with U;  PLUS==1: (I + g_mu) with U^dag (Umat preloaded transposed).
template <int MU, int PLUS>
__device__ __forceinline__ void accum_dir(const float2 psi[12], const float2 Umat[9], float2 acc[12]) {
    float2 a[3], b[3];
#pragma unroll
    for (int c = 0; c < 3; ++c) {
        float2 p0 = psi[c], p1 = psi[3 + c], p2 = psi[6 + c], p3 = psi[9 + c];
        if (MU == 0) { a[c] = PLUS ? addi(p0, p3) : subi(p0, p3); b[c] = PLUS ? addi(p1, p2) : subi(p1, p2); }
        if (MU == 1) { a[c] = PLUS ? csub(p0, p3) : cadd(p0, p3); b[c] = PLUS ? cadd(p1, p2) : csub(p1, p2); }
        if (MU == 2) { a[c] = PLUS ? addi(p0, p2) : subi(p0, p2); b[c] = PLUS ? subi(p1, p3) : addi(p1, p3); }
        if (MU == 3) { a[c] = PLUS ? cadd(p0, p2) : csub(p0, p2); b[c] = PLUS ? cadd(p1, p3) : csub(p1, p3); }
    }
#pragma unroll
    for (int C = 0; C < 3; ++C) {
        float2 sA = make_float2(0.f, 0.f), sB = make_float2(0.f, 0.f);
#pragma unroll
        for (int c = 0; c < 3; ++c) {
            float2 u = Umat[C * 3 + c];
            if (PLUS) { cfma_cj(sA, u, a[c]); cfma_cj(sB, u, b[c]); }
            else      { cfma   (sA, u, a[c]); cfma   (sB, u, b[c]); }
        }
        acc[C]     = cadd(acc[C], sA);
        acc[3 + C] = cadd(acc[3 + C], sB);
        if (MU == 0) {
            if (!PLUS) { acc[6 + C] = cadd(acc[6 + C], muli(sB));  acc[9 + C] = cadd(acc[9 + C], muli(sA)); }
            else       { acc[6 + C] = cadd(acc[6 + C], mulmi(sB)); acc[9 + C] = cadd(acc[9 + C], mulmi(sA)); }
        }
        if (MU == 1) {
            if (!PLUS) { acc[6 + C] = csub(acc[6 + C], sB); acc[9 + C] = cadd(acc[9 + C], sA); }
            else       { acc[6 + C] = cadd(acc[6 + C], sB); acc[9 + C] = csub(acc[9 + C], sA); }
        }
        if (MU == 2) {
            if (!PLUS) { acc[6 + C] = cadd(acc[6 + C], muli(sA));  acc[9 + C] = cadd(acc[9 + C], mulmi(sB)); }
            else       { acc[6 + C] = cadd(acc[6 + C], mulmi(sA)); acc[9 + C] = cadd(acc[9 + C], muli(sB)); }
        }
        if (MU == 3) {
            if (!PLUS) { acc[6 + C] = csub(acc[6 + C], sA); acc[9 + C] = csub(acc[9 + C], sB); }
            else       { acc[6 + C] = cadd(acc[6 + C], sA); acc[9 + C] = cadd(acc[9 + C], sB); }
        }
    }
}

__global__ __launch_bounds__(256) void wilson_dslash(const float2* __restrict__ src,
                                                     const float2* __restrict__ U,
                                                     float2* __restrict__ dst) {
    __shared__ float2 s_psi[12][256]; // central spinor tile: 24 KB

    const int tid = threadIdx.x;
    const unsigned site = blockIdx.x * 256u + (unsigned)tid;
    const int x = site & 31;
    const int y = (site >> 5) & 31;
    const int z = (site >> 10) & 31;
    const int t = (site >> 15) & 31;
    const int yl = tid >> 5; // wave-uniform (wave32 == one x-line)

    // --- CDNA5 async copy: stage central spinor into LDS (ASYNCcnt path) ---
    {
        const unsigned long long gb = (unsigned long long)(uintptr_t)src;
#pragma unroll
        for (int k = 0; k < 12; ++k) {
            unsigned voff = (((unsigned)k << L4_SHIFT) + site) * 8u;           // byte offset into src
            unsigned loff = (unsigned)(uintptr_t)&s_psi[k][tid];               // LDS byte offset (addr[31:0])
            asm volatile("global_load_async_to_lds_b64 %0, %1, %2"
                         :: "v"(loff), "v"(voff), "s"(gb) : "memory");
        }
    }
    asm volatile("s_wait_asynccnt 0x0" ::: "memory");
    __syncthreads();

    float2 acc[12];
#pragma unroll
    for (int k = 0; k < 12; ++k) acc[k] = make_float2(0.f, 0.f);

    float2 psi[12], Umat[9];

    // ---- mu = 0 (x): both neighbors live in the staged LDS tile ----
    {
        loadPsiL(s_psi, (tid & ~31) | ((x + 1) & 31), psi);
        loadU(U, 0, site, Umat);
        accum_dir<0, 0>(psi, Umat, acc);

        unsigned nm = (site & ~31u) | (unsigned)((x + 31) & 31);
        loadPsiL(s_psi, (tid & ~31) | ((x + 31) & 31), psi);
        loadUdag(U, 0, nm, Umat);
        accum_dir<0, 1>(psi, Umat, acc);
    }
    // ---- mu = 1 (y): interior rows hit LDS, edge rows go to global ----
    {
        unsigned np = (site & ~(31u << 5)) | ((unsigned)((y + 1) & 31) << 5);
        if (yl < 7) loadPsiL(s_psi, tid + 32, psi); else loadPsiG(src, np, psi);
        loadU(U, 1, site, Umat);
        accum_dir<1, 0>(psi, Umat, acc);

        unsigned nm = (site & ~(31u << 5)) | ((unsigned)((y + 31) & 31) << 5);
        if (yl > 0) loadPsiL(s_psi, tid - 32, psi); else loadPsiG(src, nm, psi);
        loadUdag(U, 1, nm, Umat);
        accum_dir<1, 1>(psi, Umat, acc);
    }
    // ---- mu = 2 (z) ----
    {
        unsigned np = (site & ~(31u << 10)) | ((unsigned)((z + 1) & 31) << 10);
        loadPsiG(src, np, psi);
        loadU(U, 2, site, Umat);
        accum_dir<2, 0>(psi, Umat, acc);

        unsigned nm = (site & ~(31u << 10)) | ((unsigned)((z + 31) & 31) << 10);
        loadPsiG(src, nm, psi);
        loadUdag(U, 2, nm, Umat);
        accum_dir<2, 1>(psi, Umat, acc);
    }
    // ---- mu = 3 (t) ----
    {
        unsigned np = (site & ~(31u << 15)) | ((unsigned)((t + 1) & 31) << 15);
        loadPsiG(src, np, psi);
        loadU(U, 3, site, Umat);
        accum_dir<3, 0>(psi, Umat, acc);

        unsigned nm = (site & ~(31u << 15)) | ((unsigned)((t + 31) & 31) << 15);
        loadPsiG(src, nm, psi);
        loadUdag(U, 3, nm, Umat);
        accum_dir<3, 1>(psi, Umat, acc);
    }

    // dest = psi(x) - 0.1 * acc   (central value re-read from LDS tile)
#pragma unroll
    for (int k = 0; k < 12; ++k) {
        float2 c = s_psi[k][tid];
        dst[((unsigned)k << L4_SHIFT) + site] =
            make_float2(fmaf(MKAPPA, acc[k].x, c.x), fmaf(MKAPPA, acc[k].y, c.y));
    }
}

extern "C" void kernel_launch(void* const* d_in, const int* in_sizes, int n_in,
                              void* d_out, int out_size, void* d_ws, size_t ws_size,
                              hipStream_t stream) {
    const float2* src = (const float2*)d_in[0];
    const float2* U   = (const float2*)d_in[1];
    float2* dst       = (float2*)d_out;
    (void)in_sizes; (void)n_in; (void)out_size; (void)d_ws; (void)ws_size;
    wilson_dslash<<<dim3(NBLK), dim3(256), 0, stream>>>(src, U, dst);
}